// GAT_13056700580048
// MI455X (gfx1250) — compile-verified
//
#include <hip/hip_runtime.h>

// ---------------------------------------------------------------------------
// GATv2 x2 + edge MLP for MI455X (gfx1250, wave32).
// Dense matmuls use V_WMMA_F32_16X16X4_F32 (fp32-precise CDNA5 matrix path).
// ---------------------------------------------------------------------------

typedef __attribute__((ext_vector_type(2))) float v2f;
typedef __attribute__((ext_vector_type(8))) float v8f;

#define WMMA_F32(a, b, c) \
  __builtin_amdgcn_wmma_f32_16x16x4_f32(false, (a), false, (b), (short)0, (c), false, false)

#define NEG_SLOPE 0.2f

// Monotonic float<->uint encoding for atomicMax on floats
__device__ __forceinline__ unsigned encf(float x) {
  unsigned u = __float_as_uint(x);
  return (u >> 31) ? ~u : (u | 0x80000000u);
}
__device__ __forceinline__ float decf(unsigned e) {
  return (e & 0x80000000u) ? __uint_as_float(e ^ 0x80000000u) : __uint_as_float(~e);
}

// ---------------------------------------------------------------------------
// Self-loop attribute = mean of incoming edge_attr (PyG fill_value='mean')
// ---------------------------------------------------------------------------
__global__ void k_loop_accum(const float* __restrict__ ea, const int* __restrict__ dstA,
                             float* __restrict__ loop_attr, float* __restrict__ deg, int E) {
  int g = blockIdx.x * blockDim.x + threadIdx.x;
  if (g >= E * 32) return;
  int e = g >> 5, k = g & 31;
  int d = dstA[e];
  atomicAdd(&loop_attr[(size_t)d * 32 + k], ea[(size_t)e * 32 + k]);
  if (k == 0) atomicAdd(&deg[d], 1.0f);
}

__global__ void k_loop_norm(float* __restrict__ loop_attr, const float* __restrict__ deg, int N) {
  int g = blockIdx.x * blockDim.x + threadIdx.x;
  if (g >= N * 32) return;
  int n = g >> 5;
  loop_attr[g] /= fmaxf(deg[n], 1.0f);
}

// ---------------------------------------------------------------------------
// Generic fp32 WMMA GEMM: C[M,Nc] = A[M,K] @ W[K,Nc] + bias   (row-major)
// Block = 256 threads = 8 waves, tile = 32(M) x 64(N); each wave owns a 16x16
// C-subtile. K must be a multiple of 4. A-fragment layout per ISA 7.12.2:
//   lanes 0-15 : M=lane,   VGPR0=K(k0),   VGPR1=K(k0+1)
//   lanes16-31 : M=lane-16,VGPR0=K(k0+2), VGPR1=K(k0+3)
// B-fragment mirrors it (rows K striped across lane halves).
// ---------------------------------------------------------------------------
__global__ void k_gemm_bias(const float* __restrict__ A, const float* __restrict__ W,
                            const float* __restrict__ bias, float* __restrict__ C,
                            int M, int K, int Nc) {
  int wid = threadIdx.x >> 5, lane = threadIdx.x & 31;
  int row0 = blockIdx.y * 32 + (wid >> 2) * 16;
  int col0 = blockIdx.x * 64 + (wid & 3) * 16;
  int lm = lane & 15;
  int kh = (lane >> 4) * 2;  // 0 or 2

  int r = row0 + lm;
  if (r >= M) r = M - 1;  // clamp loads; EXEC stays all-ones for WMMA
  const float* Arow = A + (size_t)r * K;

  v8f acc = {};
  for (int k0 = 0; k0 < K; k0 += 4) {
    v2f a, b;
    a.x = Arow[k0 + kh];
    a.y = Arow[k0 + kh + 1];
    b.x = W[(size_t)(k0 + kh) * Nc + col0 + lm];
    b.y = W[(size_t)(k0 + kh + 1) * Nc + col0 + lm];
    acc = WMMA_F32(a, b, acc);
  }

  int srow = row0 + ((lane >> 4) ? 8 : 0);
  int scol = col0 + lm;
  float bv = bias[scol];
#pragma unroll
  for (int j = 0; j < 8; j++) {
    int rr = srow + j;
    if (rr < M) C[(size_t)rr * Nc + scol] = acc[j] + bv;
  }
}

// ---------------------------------------------------------------------------
// Layer-1 attention logits (H=4, C=64). One wave32 per (virtual) edge.
// e >= E are self-loops. e-transform (ea[32] @ We[32,256]) computed in-wave.
// ---------------------------------------------------------------------------
__global__ void k_gat1_logits(const float* __restrict__ xl, const float* __restrict__ xr,
                              const float* __restrict__ ea, const float* __restrict__ loop_attr,
                              const int* __restrict__ srcA, const int* __restrict__ dstA,
                              const float* __restrict__ We, const float* __restrict__ att,
                              float* __restrict__ logits, unsigned* __restrict__ lmax,
                              int E, int Ep) {
  int w = blockIdx.x * 8 + (threadIdx.x >> 5);
  if (w >= Ep) return;
  int lane = threadIdx.x & 31;
  int s = (w < E) ? srcA[w] : (w - E);
  int d = (w < E) ? dstA[w] : (w - E);
  const float* eap = (w < E) ? (ea + (size_t)w * 32) : (loop_attr + (size_t)(w - E) * 32);
  float eav = eap[lane];

  const float* xls = xl + (size_t)s * 256;
  const float* xrd = xr + (size_t)d * 256;
  float acc[8];
#pragma unroll
  for (int t = 0; t < 8; t++) acc[t] = xls[t * 32 + lane] + xrd[t * 32 + lane];

#pragma unroll 4
  for (int k = 0; k < 32; k++) {
    float sv = __shfl(eav, k, 32);
#pragma unroll
    for (int t = 0; t < 8; t++) acc[t] += sv * We[k * 256 + t * 32 + lane];
  }

#pragma unroll
  for (int t = 0; t < 8; t++) {
    float v = acc[t];
    v = (v > 0.0f) ? v : NEG_SLOPE * v;
    acc[t] = v * att[t * 32 + lane];
  }
  // channel j = t*32+lane; head = j>>6  ->  head h uses t = 2h, 2h+1
  float ph[4];
#pragma unroll
  for (int h = 0; h < 4; h++) ph[h] = acc[2 * h] + acc[2 * h + 1];
#pragma unroll
  for (int off = 16; off >= 1; off >>= 1) {
#pragma unroll
    for (int h = 0; h < 4; h++) ph[h] += __shfl_xor(ph[h], off, 32);
  }
  if (lane == 0) {
#pragma unroll
    for (int h = 0; h < 4; h++) {
      logits[(size_t)w * 4 + h] = ph[h];
      atomicMax(&lmax[d * 4 + h], encf(ph[h]));
    }
  }
}

// exp(logit - max) and denominator accumulation (generic head count H)
__global__ void k_softmax(const int* __restrict__ dstA, float* __restrict__ logits,
                          const unsigned* __restrict__ lmax, float* __restrict__ denom,
                          int E, int Ep, int H) {
  int g = blockIdx.x * blockDim.x + threadIdx.x;
  if (g >= Ep * H) return;
  int e = g / H, h = g - e * H;
  int d = (e < E) ? dstA[e] : (e - E);
  float ex = __expf(logits[g] - decf(lmax[d * H + h]));
  logits[g] = ex;  // reuse buffer to hold ex
  atomicAdd(&denom[d * H + h], ex);
}

// Layer-1 aggregation: out[dst] += alpha * xl[src]   (256 channels/edge)
__global__ void k_gat1_agg(const float* __restrict__ xl, const int* __restrict__ srcA,
                           const int* __restrict__ dstA, const float* __restrict__ exv,
                           const float* __restrict__ denom, float* __restrict__ out,
                           int E, int Ep) {
  int w = blockIdx.x * 8 + (threadIdx.x >> 5);
  if (w >= Ep) return;
  int lane = threadIdx.x & 31;
  int s = (w < E) ? srcA[w] : (w - E);
  int d = (w < E) ? dstA[w] : (w - E);
  float al[4];
#pragma unroll
  for (int h = 0; h < 4; h++) al[h] = exv[(size_t)w * 4 + h] / denom[d * 4 + h];
  const float* xls = xl + (size_t)s * 256;
  float* od = out + (size_t)d * 256;
#pragma unroll
  for (int t = 0; t < 8; t++) atomicAdd(&od[t * 32 + lane], al[t >> 1] * xls[t * 32 + lane]);
}

__global__ void k_bias_relu(float* __restrict__ buf, const float* __restrict__ bias,
                            int n, int mod) {
  int g = blockIdx.x * blockDim.x + threadIdx.x;
  if (g >= n) return;
  buf[g] = fmaxf(buf[g] + bias[g % mod], 0.0f);
}

// ---------------------------------------------------------------------------
// Layer-2 attention (H=1, C=64), one wave per edge
// ---------------------------------------------------------------------------
__global__ void k_gat2_logits(const float* __restrict__ xl, const float* __restrict__ xr,
                              const float* __restrict__ ea, const float* __restrict__ loop_attr,
                              const int* __restrict__ srcA, const int* __restrict__ dstA,
                              const float* __restrict__ We, const float* __restrict__ att,
                              float* __restrict__ logits, unsigned* __restrict__ lmax,
                              int E, int Ep) {
  int w = blockIdx.x * 8 + (threadIdx.x >> 5);
  if (w >= Ep) return;
  int lane = threadIdx.x & 31;
  int s = (w < E) ? srcA[w] : (w - E);
  int d = (w < E) ? dstA[w] : (w - E);
  const float* eap = (w < E) ? (ea + (size_t)w * 32) : (loop_attr + (size_t)(w - E) * 32);
  float eav = eap[lane];

  float a0 = xl[(size_t)s * 64 + lane] + xr[(size_t)d * 64 + lane];
  float a1 = xl[(size_t)s * 64 + 32 + lane] + xr[(size_t)d * 64 + 32 + lane];
#pragma unroll 4
  for (int k = 0; k < 32; k++) {
    float sv = __shfl(eav, k, 32);
    a0 += sv * We[k * 64 + lane];
    a1 += sv * We[k * 64 + 32 + lane];
  }
  a0 = ((a0 > 0.0f) ? a0 : NEG_SLOPE * a0) * att[lane];
  a1 = ((a1 > 0.0f) ? a1 : NEG_SLOPE * a1) * att[32 + lane];
  float p = a0 + a1;
#pragma unroll
  for (int off = 16; off >= 1; off >>= 1) p += __shfl_xor(p, off, 32);
  if (lane == 0) {
    logits[w] = p;
    atomicMax(&lmax[d], encf(p));
  }
}

__global__ void k_gat2_agg(const float* __restrict__ xl, const int* __restrict__ srcA,
                           const int* __restrict__ dstA, const float* __restrict__ exv,
                           const float* __restrict__ denom, float* __restrict__ out,
                           int E, int Ep) {
  int w = blockIdx.x * 8 + (threadIdx.x >> 5);
  if (w >= Ep) return;
  int lane = threadIdx.x & 31;
  int s = (w < E) ? srcA[w] : (w - E);
  int d = (w < E) ? dstA[w] : (w - E);
  float al = exv[w] / denom[d];
  atomicAdd(&out[(size_t)d * 64 + lane], al * xl[(size_t)s * 64 + lane]);
  atomicAdd(&out[(size_t)d * 64 + 32 + lane], al * xl[(size_t)s * 64 + 32 + lane]);
}

// ---------------------------------------------------------------------------
// Fused edge MLP: 16 edges/block.  repr[16x160] -> WMMA GEMM -> relu ->
// WMMA GEMM -> row L2-normalize -> out[16x128].
// ---------------------------------------------------------------------------
__global__ void k_edge_mlp(const float* __restrict__ h2, const float* __restrict__ ea,
                           const int* __restrict__ srcA, const int* __restrict__ dstA,
                           const float* __restrict__ Pw1, const float* __restrict__ pb1,
                           const float* __restrict__ Pw2, const float* __restrict__ pb2,
                           float* __restrict__ out, int E) {
  __shared__ float repr[16 * 160];
  __shared__ float hid[16 * 128];
  __shared__ float emb[16 * 128];
  __shared__ float norms[16];

  int tid = threadIdx.x;
  int ebase = blockIdx.x * 16;

  for (int idx = tid; idx < 16 * 160; idx += 256) {
    int r = idx / 160, c = idx - r * 160;
    int e = ebase + r;
    if (e >= E) e = E - 1;  // clamp (E is a multiple of 16, defensive)
    float v;
    if (c < 64)       v = h2[(size_t)srcA[e] * 64 + c];
    else if (c < 128) v = h2[(size_t)dstA[e] * 64 + (c - 64)];
    else              v = ea[(size_t)e * 32 + (c - 128)];
    repr[idx] = v;
  }
  __syncthreads();

  int wid = tid >> 5, lane = tid & 31;
  int lm = lane & 15, kh = (lane >> 4) * 2;
  int col0 = wid * 16;
  int srow = (lane >> 4) ? 8 : 0;

  // GEMM 1: [16x160] @ Pw1[160,128] + pb1, relu
  v8f acc = {};
  for (int k0 = 0; k0 < 160; k0 += 4) {
    v2f a, b;
    a.x = repr[lm * 160 + k0 + kh];
    a.y = repr[lm * 160 + k0 + kh + 1];
    b.x = Pw1[(k0 + kh) * 128 + col0 + lm];
    b.y = Pw1[(k0 + kh + 1) * 128 + col0 + lm];
    acc = WMMA_F32(a, b, acc);
  }
  {
    float bv = pb1[col0 + lm];
#pragma unroll
    for (int j = 0; j < 8; j++) hid[(srow + j) * 128 + col0 + lm] = fmaxf(acc[j] + bv, 0.0f);
  }
  __syncthreads();

  // GEMM 2: hid[16x128] @ Pw2[128,128] + pb2
  v8f acc2 = {};
  for (int k0 = 0; k0 < 128; k0 += 4) {
    v2f a, b;
    a.x = hid[lm * 128 + k0 + kh];
    a.y = hid[lm * 128 + k0 + kh + 1];
    b.x = Pw2[(k0 + kh) * 128 + col0 + lm];
    b.y = Pw2[(k0 + kh + 1) * 128 + col0 + lm];
    acc2 = WMMA_F32(a, b, acc2);
  }
  {
    float bv = pb2[col0 + lm];
#pragma unroll
    for (int j = 0; j < 8; j++) emb[(srow + j) * 128 + col0 + lm] = acc2[j] + bv;
  }
  __syncthreads();

  if (tid < 16) {
    float ssum = 0.0f;
    for (int c = 0; c < 128; c++) { float v = emb[tid * 128 + c]; ssum += v * v; }
    norms[tid] = fmaxf(sqrtf(ssum), 1e-12f);
  }
  __syncthreads();

  for (int idx = tid; idx < 16 * 128; idx += 256) {
    int r = idx >> 7;
    if (ebase + r < E) out[(size_t)ebase * 128 + idx] = emb[idx] / norms[r];
  }
}

// ---------------------------------------------------------------------------
// Host orchestration
// ---------------------------------------------------------------------------
extern "C" void kernel_launch(void* const* d_in, const int* in_sizes, int n_in,
                              void* d_out, int out_size, void* d_ws, size_t ws_size,
                              hipStream_t stream) {
  const float* x     = (const float*)d_in[0];
  const float* ea    = (const float*)d_in[1];
  const int*   ei    = (const int*)d_in[2];
  const float* Wl1   = (const float*)d_in[3];
  const float* bl1   = (const float*)d_in[4];
  const float* Wr1   = (const float*)d_in[5];
  const float* br1   = (const float*)d_in[6];
  const float* We1   = (const float*)d_in[7];
  const float* att1  = (const float*)d_in[8];
  const float* bias1 = (const float*)d_in[9];
  const float* Wl2   = (const float*)d_in[10];
  const float* bl2   = (const float*)d_in[11];
  const float* Wr2   = (const float*)d_in[12];
  const float* br2   = (const float*)d_in[13];
  const float* We2   = (const float*)d_in[14];
  const float* att2  = (const float*)d_in[15];
  const float* bias2 = (const float*)d_in[16];
  const float* Pw1   = (const float*)d_in[17];
  const float* pb1   = (const float*)d_in[18];
  const float* Pw2   = (const float*)d_in[19];
  const float* pb2   = (const float*)d_in[20];

  const int N  = in_sizes[0] / 128;
  const int E  = in_sizes[1] / 32;
  const int Ep = E + N;
  const int* srcA = ei;
  const int* dstA = ei + E;

  // ---- workspace carve-up (floats) ----
  float* ws = (float*)d_ws;
  float*    xl1       = ws;                              // N*256
  float*    xr1       = xl1 + (size_t)N * 256;           // N*256
  float*    out1      = xr1 + (size_t)N * 256;           // N*256  (becomes h1)
  float*    loop_attr = out1 + (size_t)N * 256;          // N*32
  float*    deg       = loop_attr + (size_t)N * 32;      // N
  unsigned* lmax1     = (unsigned*)(deg + N);            // N*4
  float*    denom1    = (float*)lmax1 + (size_t)N * 4;   // N*4
  float*    logits1   = denom1 + (size_t)N * 4;          // Ep*4
  float*    xl2       = logits1 + (size_t)Ep * 4;        // N*64
  float*    xr2       = xl2 + (size_t)N * 64;            // N*64
  float*    out2      = xr2 + (size_t)N * 64;            // N*64  (becomes h2)
  unsigned* lmax2     = (unsigned*)(out2 + (size_t)N * 64); // N
  float*    denom2    = (float*)lmax2 + N;               // N
  float*    logits2   = denom2 + N;                      // Ep

  // ---- zero accumulators (lmax encoded-min == 0) ----
  hipMemsetAsync(out1, 0, (size_t)N * 256 * 4, stream);
  hipMemsetAsync(loop_attr, 0, (size_t)N * 33 * 4, stream);       // loop_attr + deg
  hipMemsetAsync(lmax1, 0, (size_t)N * 8 * 4, stream);            // lmax1 + denom1
  hipMemsetAsync(out2, 0, (size_t)N * 64 * 4, stream);
  hipMemsetAsync(lmax2, 0, (size_t)N * 2 * 4, stream);            // lmax2 + denom2

  const int T = 256;
  // ---- self-loop attrs ----
  k_loop_accum<<<(E * 32 + T - 1) / T, T, 0, stream>>>(ea, dstA, loop_attr, deg, E);
  k_loop_norm<<<(N * 32 + T - 1) / T, T, 0, stream>>>(loop_attr, deg, N);

  // ---- layer 1: node transforms (WMMA GEMM) ----
  dim3 g1(4, (N + 31) / 32);
  k_gemm_bias<<<g1, T, 0, stream>>>(x, Wl1, bl1, xl1, N, 128, 256);
  k_gemm_bias<<<g1, T, 0, stream>>>(x, Wr1, br1, xr1, N, 128, 256);

  int eblk = (Ep + 7) / 8;  // 8 waves/block, 1 wave per edge
  k_gat1_logits<<<eblk, T, 0, stream>>>(xl1, xr1, ea, loop_attr, srcA, dstA, We1, att1,
                                        logits1, lmax1, E, Ep);
  k_softmax<<<(Ep * 4 + T - 1) / T, T, 0, stream>>>(dstA, logits1, lmax1, denom1, E, Ep, 4);
  k_gat1_agg<<<eblk, T, 0, stream>>>(xl1, srcA, dstA, logits1, denom1, out1, E, Ep);
  k_bias_relu<<<((size_t)N * 256 + T - 1) / T, T, 0, stream>>>(out1, bias1, N * 256, 256);

  // ---- layer 2 ----
  dim3 g2(1, (N + 31) / 32);
  k_gemm_bias<<<g2, T, 0, stream>>>(out1, Wl2, bl2, xl2, N, 256, 64);
  k_gemm_bias<<<g2, T, 0, stream>>>(out1, Wr2, br2, xr2, N, 256, 64);

  k_gat2_logits<<<eblk, T, 0, stream>>>(xl2, xr2, ea, loop_attr, srcA, dstA, We2, att2,
                                        logits2, lmax2, E, Ep);
  k_softmax<<<(Ep + T - 1) / T, T, 0, stream>>>(dstA, logits2, lmax2, denom2, E, Ep, 1);
  k_gat2_agg<<<eblk, T, 0, stream>>>(xl2, srcA, dstA, logits2, denom2, out2, E, Ep);
  k_bias_relu<<<((size_t)N * 64 + T - 1) / T, T, 0, stream>>>(out2, bias2, N * 64, 64);

  // ---- fused edge MLP + normalize (WMMA) ----
  k_edge_mlp<<<(E + 15) / 16, T, 0, stream>>>(out2, ea, srcA, dstA, Pw1, pb1, Pw2, pb2,
                                              (float*)d_out, E);
}